// TransportOperator_18966575579563
// MI455X (gfx1250) — compile-verified
//
#include <hip/hip_runtime.h>

// TransportOperator: y = expm(sum_m c[b,m] * psi[m,k]) @ z_block, 3 transports.
// Fused Taylor series y = sum_j A^j z / j!  (||A||~0.05 -> 12 terms >> f32 precision).
// Workgroup: one k (0..7), 32 samples, one transport; 4 wave32.
// Per term:  W = sum_m P_mk x (V * c[:,m])  via V_WMMA_F32_16X16X4_F32 (pure f32).
// Each wave: one 16-row strip x TWO 16-col sample tiles -> each psi A-fragment
// load feeds 2 wmma (halves L2 psi traffic vs TB=16) and the acc0/acc1
// alternation breaks the WMMA accumulator RAW chain.

typedef __attribute__((ext_vector_type(2))) float v2f;
typedef __attribute__((ext_vector_type(8))) float v8f;

#define Bn 1024
#define Dn 512
#define Mn 16
#define Kn 8
#define Nn 64
#define TB 32        // samples per workgroup (2 column tiles per wave)
#define VS 44        // LDS row stride: 2-row delta -> bank+24, 8-row delta -> bank+32 (conflict-free)
#define JTERMS 12

__global__ __launch_bounds__(128) void transport_taylor_wmma(
    const float* __restrict__ z0, const float* __restrict__ z1,
    const float* __restrict__ c_enc, const float* __restrict__ c0p,
    const float* __restrict__ c1p, const float* __restrict__ psi,
    float* __restrict__ out)
{
    const int k  = blockIdx.x;            // block index 0..7
    const int b0 = blockIdx.y * TB;       // sample tile base
    const int t  = blockIdx.z;            // transport 0,1,2

    const float* c = (t == 0) ? c_enc : (t == 1) ? c0p : c1p;
    const float* z = (t == 2) ? z1 : z0;
    float* o = out + (size_t)t * Bn * Dn;

    __shared__ float cl[Mn][TB];          // cl[m][b_local]
    __shared__ float V[Nn][VS];           // current Taylor vector, column = sample (cols 0..31 used)

    const int tid  = threadIdx.x;
    const int wave = tid >> 5;            // 0..3  (wave32)
    const int lane = tid & 31;
    const int bl   = lane & 15;           // WMMA N-dim column within a tile
    const int half = lane >> 4;           // lane half -> K/M sub-pair per ISA layout

    // Stage coefficients: cl[m][b] = c[(b0+b)*M + m]
    for (int f = tid; f < Mn * TB; f += 128) {
        int bb = f >> 4, m = f & 15;
        cl[m][bb] = c[(size_t)(b0 + bb) * Mn + m];
    }
    // Stage z block: V[v][b] = z[(b0+b)*D + k*64 + v]   (4 threads x 16 floats per sample)
    {
        int s = tid >> 2, sg = tid & 3;
        const float* zp = z + (size_t)(b0 + s) * Dn + k * Nn + sg * 16;
        #pragma unroll
        for (int i = 0; i < 16; ++i) V[sg * 16 + i][s] = zp[i];
    }
    __syncthreads();

    // y accumulators = v_0 fragments (C/D layout: row = 16*wave + 8*half + r, col = ct*16 + bl)
    v8f y0, y1;
    #pragma unroll
    for (int r = 0; r < 8; ++r) {
        y0[r] = V[16 * wave + 8 * half + r][bl];
        y1[r] = V[16 * wave + 8 * half + r][16 + bl];
    }

    const float* psik = psi + (size_t)k * Nn * Nn;   // + m*K*N*N selects psi[m][k]

    for (int j = 1; j <= JTERMS; ++j) {
        v8f acc0 = {}, acc1 = {};
        #pragma unroll 1
        for (int m = 0; m < Mn; ++m) {
            const float cm0 = cl[m][bl];
            const float cm1 = cl[m][16 + bl];
            // A-operand rows for this wave/lane: u = 16*wave + bl
            const float* P = psik + (size_t)m * (Kn * Nn * Nn)
                                  + (size_t)(16 * wave + bl) * Nn;
            #pragma unroll
            for (int kk = 0; kk < 16; ++kk) {       // K = 64 in steps of 4
                const int kb = kk * 4 + half * 2;
                v2f a;  a[0]  = P[kb];                 a[1]  = P[kb + 1];   // 16x4 f32 A layout
                const float u0 = V[kb][bl],       u1 = V[kb + 1][bl];
                const float w0 = V[kb][16 + bl],  w1 = V[kb + 1][16 + bl];
                v2f bB; bB[0] = u0 * cm0;  bB[1] = u1 * cm0;               // 4x16 B, c folded in
                v2f bC; bC[0] = w0 * cm1;  bC[1] = w1 * cm1;
                acc0 = __builtin_amdgcn_wmma_f32_16x16x4_f32(
                           false, a, false, bB, (short)0, acc0, false, false);
                acc1 = __builtin_amdgcn_wmma_f32_16x16x4_f32(
                           false, a, false, bC, (short)0, acc1, false, false);
            }
        }
        const float inv = 1.0f / (float)j;
        __syncthreads();                     // all waves done reading V (v_{j-1})
        #pragma unroll
        for (int r = 0; r < 8; ++r) {
            const float vj0 = acc0[r] * inv; // v_j = A v_{j-1} / j
            const float vj1 = acc1[r] * inv;
            y0[r] += vj0;
            y1[r] += vj1;
            V[16 * wave + 8 * half + r][bl]      = vj0;
            V[16 * wave + 8 * half + r][16 + bl] = vj1;
        }
        __syncthreads();
    }

    // Write y via LDS so global stores are contiguous per sample
    #pragma unroll
    for (int r = 0; r < 8; ++r) {
        V[16 * wave + 8 * half + r][bl]      = y0[r];
        V[16 * wave + 8 * half + r][16 + bl] = y1[r];
    }
    __syncthreads();
    {
        int s = tid >> 2, sg = tid & 3;
        float* op = o + (size_t)(b0 + s) * Dn + k * Nn + sg * 16;
        #pragma unroll
        for (int i = 0; i < 16; ++i) op[i] = V[sg * 16 + i][s];
    }
}

extern "C" void kernel_launch(void* const* d_in, const int* in_sizes, int n_in,
                              void* d_out, int out_size, void* d_ws, size_t ws_size,
                              hipStream_t stream) {
    const float* z0  = (const float*)d_in[0];
    const float* z1  = (const float*)d_in[1];
    const float* ce  = (const float*)d_in[2];
    const float* c0  = (const float*)d_in[3];
    const float* c1  = (const float*)d_in[4];
    const float* psi = (const float*)d_in[5];
    float* out = (float*)d_out;

    dim3 grid(Kn, Bn / TB, 3);   // 8 x 32 x 3 = 768 workgroups
    transport_taylor_wmma<<<grid, 128, 0, stream>>>(z0, z1, ce, c0, c1, psi, out);
}